// Attention_38860864094288
// MI455X (gfx1250) — compile-verified
//
#include <hip/hip_runtime.h>
#include <hip/hip_bf16.h>
#include <math.h>

// B=2, N=512, C=256, fp32 end-to-end.
// Pipeline: (1) q,k,v = x@W^T via V_WMMA_F32_16X16X4_F32
//           (2) score+softmax (pure VALU, no matmul structure)
//           (3) o = attn@v via WMMA
//           (4) out = o@Wp^T + bp via WMMA

typedef float v2f __attribute__((ext_vector_type(2)));
typedef float v8f __attribute__((ext_vector_type(8)));

#define DIM_B 2
#define DIM_N 512
#define DIM_C 256

// ---------------------------------------------------------------------------
// Y[M x 256] = X[M x 256] @ W^T (+ optional bias).  One wave per 16x16 tile.
// grid = (M/16, 256/16), block = 32.
// ---------------------------------------------------------------------------
__global__ __launch_bounds__(32)
void gemm_xwT_f32(const float* __restrict__ X,
                  const float* __restrict__ W,
                  const float* __restrict__ bias,
                  float* __restrict__ Y) {
    const int C = DIM_C;
    const int mBase = blockIdx.x << 4;
    const int nBase = blockIdx.y << 4;
    const int lane  = threadIdx.x;
    const int half  = lane >> 4;      // 0: K pair {0,1}, 1: K pair {2,3}
    const int l16   = lane & 15;

    const float* __restrict__ arow = X + (size_t)(mBase + l16) * C;   // A: M = l16
    const float* __restrict__ brow = W + (size_t)(nBase + l16) * C;   // B[k][n] = W[n][k]

    v8f acc = {};
    for (int kk = 0; kk < C; kk += 4) {
        // A[16x4]: lane holds A[m][2*half + {0,1}]  -> contiguous 8B load
        v2f a = *(const v2f*)(arow + kk + 2 * half);
        // B[4x16]: vgpr v holds B[v + 2*half][n] = W[n][kk + v + 2*half] -> contiguous 8B load
        v2f b = *(const v2f*)(brow + kk + 2 * half);
        acc = __builtin_amdgcn_wmma_f32_16x16x4_f32(
            /*neg_a=*/false, a, /*neg_b=*/false, b,
            /*c_mod=*/(short)0, acc, /*reuse_a=*/false, /*reuse_b=*/false);
    }

    const float bval = bias ? bias[nBase + l16] : 0.0f;
#pragma unroll
    for (int r = 0; r < 8; ++r) {
        const int row = mBase + r + 8 * half;   // C/D: vgpr r -> M = r + 8*half
        Y[(size_t)row * C + nBase + l16] = acc[r] + bval;
    }
}

// ---------------------------------------------------------------------------
// score[b,i,j] = sum_c relu(q[b,i,c] + k[b,j,c]); softmax over j -> attn.
// One block of 256 threads per (b,i) row.  grid = B*N.
// ---------------------------------------------------------------------------
__global__ __launch_bounds__(256)
void score_softmax(const float* __restrict__ q,
                   const float* __restrict__ k,
                   float* __restrict__ attn) {
    __shared__ float qs[DIM_C];
    __shared__ float sc[DIM_N];
    __shared__ float red[8];

    const int bi = blockIdx.x;            // b*N + i
    const int b  = bi >> 9;
    const int t  = threadIdx.x;

    qs[t] = q[(size_t)bi * DIM_C + t];
    __syncthreads();

    const float* __restrict__ kb = k + (size_t)b * DIM_N * DIM_C;
#pragma unroll
    for (int jj = 0; jj < 2; ++jj) {
        const int j = t + jj * 256;
        const float* __restrict__ krow = kb + (size_t)j * DIM_C;
        float s = 0.0f;
        for (int c = 0; c < DIM_C; c += 4) {
            float4 kv = *(const float4*)(krow + c);
            float4 qv = *(const float4*)(&qs[c]);
            s += fmaxf(qv.x + kv.x, 0.0f) + fmaxf(qv.y + kv.y, 0.0f) +
                 fmaxf(qv.z + kv.z, 0.0f) + fmaxf(qv.w + kv.w, 0.0f);
        }
        sc[j] = s;
    }
    __syncthreads();

    // row max over 512 values (each thread owns 2)
    const float s0 = sc[t];
    const float s1 = sc[t + 256];
    float m = fmaxf(s0, s1);
#pragma unroll
    for (int off = 16; off > 0; off >>= 1)
        m = fmaxf(m, __shfl_xor(m, off, 32));
    if ((t & 31) == 0) red[t >> 5] = m;
    __syncthreads();
    float gm = red[0];
#pragma unroll
    for (int w = 1; w < 8; ++w) gm = fmaxf(gm, red[w]);

    // exp + sum
    const float e0 = __expf(s0 - gm);
    const float e1 = __expf(s1 - gm);
    float ss = e0 + e1;
#pragma unroll
    for (int off = 16; off > 0; off >>= 1)
        ss += __shfl_xor(ss, off, 32);
    __syncthreads();
    if ((t & 31) == 0) red[t >> 5] = ss;
    __syncthreads();
    float gs = 0.0f;
#pragma unroll
    for (int w = 0; w < 8; ++w) gs += red[w];
    const float inv = 1.0f / gs;

    float* __restrict__ arow = attn + (size_t)bi * DIM_N;
    arow[t]       = e0 * inv;
    arow[t + 256] = e1 * inv;
}

// ---------------------------------------------------------------------------
// o[b] = attn[b] (512x512) @ v[b] (512x256) via f32 WMMA.
// grid = (512/16, 256/16, B), block = 32.
// ---------------------------------------------------------------------------
__global__ __launch_bounds__(32)
void attn_v_f32(const float* __restrict__ attn,
                const float* __restrict__ v,
                float* __restrict__ o) {
    const int b = blockIdx.z;
    const float* __restrict__ A = attn + (size_t)b * DIM_N * DIM_N;
    const float* __restrict__ V = v    + (size_t)b * DIM_N * DIM_C;

    const int mBase = blockIdx.x << 4;
    const int nBase = blockIdx.y << 4;
    const int lane  = threadIdx.x;
    const int half  = lane >> 4;
    const int l16   = lane & 15;

    const float* __restrict__ arow = A + (size_t)(mBase + l16) * DIM_N;

    v8f acc = {};
    for (int kk = 0; kk < DIM_N; kk += 4) {
        v2f a = *(const v2f*)(arow + kk + 2 * half);
        const int kb = kk + 2 * half;
        v2f bb;
        bb.x = V[(size_t)(kb + 0) * DIM_C + nBase + l16];  // B[k][n] = v[b,k,n]
        bb.y = V[(size_t)(kb + 1) * DIM_C + nBase + l16];
        acc = __builtin_amdgcn_wmma_f32_16x16x4_f32(
            false, a, false, bb, (short)0, acc, false, false);
    }

#pragma unroll
    for (int r = 0; r < 8; ++r) {
        const int row = mBase + r + 8 * half;
        o[((size_t)b * DIM_N + row) * DIM_C + nBase + l16] = acc[r];
    }
}

// ---------------------------------------------------------------------------
extern "C" void kernel_launch(void* const* d_in, const int* in_sizes, int n_in,
                              void* d_out, int out_size, void* d_ws, size_t ws_size,
                              hipStream_t stream) {
    const float* x  = (const float*)d_in[0];
    const float* Wq = (const float*)d_in[1];
    const float* Wk = (const float*)d_in[2];
    const float* Wv = (const float*)d_in[3];
    const float* Wp = (const float*)d_in[4];
    const float* bp = (const float*)d_in[5];
    float* out = (float*)d_out;

    float* ws = (float*)d_ws;
    const size_t MC = (size_t)DIM_B * DIM_N * DIM_C;   // 262144
    const size_t NN = (size_t)DIM_B * DIM_N * DIM_N;   // 524288
    float* q    = ws;
    float* k    = ws + MC;
    float* v    = ws + 2 * MC;
    float* attn = ws + 3 * MC;
    float* o    = attn + NN;

    const dim3 gGemm((DIM_B * DIM_N) / 16, DIM_C / 16);         // (64, 16)
    gemm_xwT_f32<<<gGemm, 32, 0, stream>>>(x, Wq, nullptr, q);
    gemm_xwT_f32<<<gGemm, 32, 0, stream>>>(x, Wk, nullptr, k);
    gemm_xwT_f32<<<gGemm, 32, 0, stream>>>(x, Wv, nullptr, v);

    score_softmax<<<DIM_B * DIM_N, 256, 0, stream>>>(q, k, attn);

    const dim3 gAV(DIM_N / 16, DIM_C / 16, DIM_B);              // (32, 16, 2)
    attn_v_f32<<<gAV, 32, 0, stream>>>(attn, v, o);

    gemm_xwT_f32<<<gGemm, 32, 0, stream>>>(o, Wp, bp, out);
}